// SparseConvBlock3D_30253749633579
// MI455X (gfx1250) — compile-verified
//
#include <hip/hip_runtime.h>
#include <hip/hip_bf16.h>

typedef float v2f __attribute__((ext_vector_type(2)));
typedef float v8f __attribute__((ext_vector_type(8)));

#define BB   8
#define NVOX 65536
#define CIN  16
#define GSZ  64
#define G3   262144
#define MAPSTRIDE (G3 + 1)   // +1 sentinel (-1) per batch for OOB neighbors
#define BNEPS 1e-3f

// ---------------------------------------------------------------------------
// Kernel 1: reset index map (incl. sentinels) to -1, zero BN stats + zero-row.
// Re-run every launch -> deterministic across graph replays.
// ---------------------------------------------------------------------------
__global__ __launch_bounds__(256)
void sc3d_init(int* __restrict__ imap, float* __restrict__ statszone) {
  int i = blockIdx.x * 256 + threadIdx.x;
  if (i < BB * MAPSTRIDE) imap[i] = -1;
  if (i < 80) statszone[i] = 0.0f;   // stats[64] + zrow[16]
}

// ---------------------------------------------------------------------------
// Kernel 2: scatter valid voxel row-indices into the dense (B, 64^3) map.
// ---------------------------------------------------------------------------
__global__ __launch_bounds__(256)
void sc3d_scatter(int* __restrict__ imap, const int* __restrict__ xyz,
                  const int* __restrict__ nvalid) {
  int i = blockIdx.x * 256 + threadIdx.x;
  if (i >= BB * NVOX) return;
  int b = i >> 16;
  int n = i & (NVOX - 1);
  if (n < nvalid[b]) {
    const int* c = xyz + (size_t)i * 3;
    int lin = (c[0] << 12) | (c[1] << 6) | c[2];
    imap[(size_t)b * MAPSTRIDE + lin] = n;
  }
}

// ---------------------------------------------------------------------------
// Kernel 3: submanifold sparse conv via V_WMMA_F32_16X16X4_F32.
// One wave32 = two 16-voxel tiles (32 voxels): B-fragments from LDS are shared
// by both accumulators (8 WMMAs per offset for 8 ds loads). Offset loop fully
// unrolled so dx/dy/dz fold to constants. Gathers are branchless: OOB -> map
// sentinel (-1), inactive -> zero-row, via address cndmasks only.
// ---------------------------------------------------------------------------
__global__ __launch_bounds__(256)
void sc3d_conv(const float* __restrict__ feats, const int* __restrict__ xyz,
               const int* __restrict__ nvalid, const float* __restrict__ w,
               const float* __restrict__ nw, const int* __restrict__ imap,
               const float* __restrict__ zrow,
               float* __restrict__ out, float* __restrict__ stats) {
  __shared__ float lw[27 * 16 * 16];
  __shared__ float red[32];

  int tid = threadIdx.x;
  for (int i = tid; i < 27 * 256; i += 256) lw[i] = w[i];
  if (tid < 32) red[tid] = 0.0f;
  __syncthreads();

  int wave = tid >> 5;
  int lane = tid & 31;
  int pair = blockIdx.x * 8 + wave;          // 0 .. B*N/32-1
  int b    = pair >> 11;                     // N/32 = 2048 pairs per batch
  int m0   = (pair & 2047) << 5;             // first of 32 voxels
  int col  = lane & 15;
  int half = lane >> 4;                      // 0: K lo / rows 0-7, 1: K hi / rows 8-15
  int klo  = half * 2;

  const int* cp0 = xyz + ((size_t)b * NVOX + (m0 + col)) * 3;
  const int* cp1 = cp0 + 16 * 3;
  int cx0 = cp0[0], cy0 = cp0[1], cz0 = cp0[2];
  int cx1 = cp1[0], cy1 = cp1[1], cz1 = cp1[2];
  int nv = nvalid[b];

  const float* fb   = feats + (size_t)b * NVOX * CIN;
  const int*   bmap = imap + (size_t)b * MAPSTRIDE;
  const float* zk   = zrow + klo;

  v8f acc0 = {0.f, 0.f, 0.f, 0.f, 0.f, 0.f, 0.f, 0.f};
  v8f acc1 = {0.f, 0.f, 0.f, 0.f, 0.f, 0.f, 0.f, 0.f};
  float ns0 = 0.0f, ns1 = 0.0f;

  #pragma unroll
  for (int o = 0; o < 27; ++o) {
    const int dx = o / 9 - 1, dy = (o / 3) % 3 - 1, dz = o % 3 - 1;
    float nwo = nw[o];

    // ---- tile 0 gather -------------------------------------------------
    int nx = cx0 + dx, ny = cy0 + dy, nz = cz0 + dz;
    bool inb = (unsigned)(nx | ny | nz) < 64u;
    unsigned lin = inb ? (((unsigned)nx << 12) | ((unsigned)ny << 6) | (unsigned)nz)
                       : (unsigned)G3;
    int nidx = bmap[lin];
    bool ok0 = nidx >= 0;
    const float* s0 = ok0 ? (fb + (size_t)nidx * CIN + klo) : zk;
    v2f a0 = *(const v2f*)(s0 + 0);
    v2f a1 = *(const v2f*)(s0 + 4);
    v2f a2 = *(const v2f*)(s0 + 8);
    v2f a3 = *(const v2f*)(s0 + 12);
    ns0 += ok0 ? nwo : 0.0f;

    // ---- tile 1 gather -------------------------------------------------
    nx = cx1 + dx; ny = cy1 + dy; nz = cz1 + dz;
    inb = (unsigned)(nx | ny | nz) < 64u;
    lin = inb ? (((unsigned)nx << 12) | ((unsigned)ny << 6) | (unsigned)nz)
              : (unsigned)G3;
    nidx = bmap[lin];
    bool ok1 = nidx >= 0;
    const float* s1 = ok1 ? (fb + (size_t)nidx * CIN + klo) : zk;
    v2f e0 = *(const v2f*)(s1 + 0);
    v2f e1 = *(const v2f*)(s1 + 4);
    v2f e2 = *(const v2f*)(s1 + 8);
    v2f e3 = *(const v2f*)(s1 + 12);
    ns1 += ok1 ? nwo : 0.0f;

    // ---- B fragments (shared by both tiles) ----------------------------
    const float* wo = lw + o * 256 + col;    // lw[o][k][n], stride 16 over k
    v2f b0 = { wo[(0  + klo) * 16], wo[(1  + klo) * 16] };
    v2f b1 = { wo[(4  + klo) * 16], wo[(5  + klo) * 16] };
    v2f b2 = { wo[(8  + klo) * 16], wo[(9  + klo) * 16] };
    v2f b3 = { wo[(12 + klo) * 16], wo[(13 + klo) * 16] };

    acc0 = __builtin_amdgcn_wmma_f32_16x16x4_f32(false, a0, false, b0, (short)0, acc0, false, false);
    acc1 = __builtin_amdgcn_wmma_f32_16x16x4_f32(false, e0, false, b0, (short)0, acc1, false, false);
    acc0 = __builtin_amdgcn_wmma_f32_16x16x4_f32(false, a1, false, b1, (short)0, acc0, false, false);
    acc1 = __builtin_amdgcn_wmma_f32_16x16x4_f32(false, e1, false, b1, (short)0, acc1, false, false);
    acc0 = __builtin_amdgcn_wmma_f32_16x16x4_f32(false, a2, false, b2, (short)0, acc0, false, false);
    acc1 = __builtin_amdgcn_wmma_f32_16x16x4_f32(false, e2, false, b2, (short)0, acc1, false, false);
    acc0 = __builtin_amdgcn_wmma_f32_16x16x4_f32(false, a3, false, b3, (short)0, acc0, false, false);
    acc1 = __builtin_amdgcn_wmma_f32_16x16x4_f32(false, e3, false, b3, (short)0, acc1, false, false);
  }

  // D layout: acc[r], lanes 0-15 -> voxel r, lanes 16-31 -> voxel r+8 (within
  // the tile), channel = lane&15. normsum for voxel V lives in lane V (& V+16).
  float lsum = 0.0f, lsq = 0.0f;
  float* ob = out + ((size_t)b * NVOX + m0) * CIN + col;
  #pragma unroll
  for (int r = 0; r < 8; ++r) {
    int mv = r + half * 8;
    float sc0 = 1.0f / (1.0f + fabsf(__shfl(ns0, mv, 32)));
    float v0  = ((m0 + mv) < nv) ? acc0[r] * sc0 : 0.0f;
    ob[(size_t)mv * CIN] = v0;
    float sc1 = 1.0f / (1.0f + fabsf(__shfl(ns1, mv, 32)));
    float v1  = ((m0 + 16 + mv) < nv) ? acc1[r] * sc1 : 0.0f;
    ob[(size_t)(16 + mv) * CIN] = v1;
    lsum += v0 + v1;
    lsq  += v0 * v0 + v1 * v1;
  }
  atomicAdd(&red[col], lsum);
  atomicAdd(&red[16 + col], lsq);
  __syncthreads();
  if (tid < 32) atomicAdd(&stats[tid], red[tid]);
}

// ---------------------------------------------------------------------------
// Kernel 4: masked batch-norm + ReLU, in place, float4 per thread.
// var = E[x^2] - mean^2 over valid entries (masked entries contribute 0 to
// the sums and are excluded by cnt), matching the reference's masked_bn.
// ---------------------------------------------------------------------------
__global__ __launch_bounds__(256)
void sc3d_bn(float* __restrict__ data, const int* __restrict__ nvalid,
             const float* __restrict__ gamma, const float* __restrict__ beta,
             const float* __restrict__ stats) {
  int i = blockIdx.x * 256 + threadIdx.x;          // over B*N*4 quads
  if (i >= BB * NVOX * 4) return;
  int cq = (i & 3) * 4;                            // channel base (0,4,8,12)
  int bn = i >> 2;
  int b = bn >> 16;
  int n = bn & (NVOX - 1);

  float cnt = 0.0f;
  #pragma unroll
  for (int k = 0; k < BB; ++k) cnt += (float)nvalid[k];
  bool valid = n < nvalid[b];

  float4 x = ((const float4*)data)[i];
  float y[4] = {x.x, x.y, x.z, x.w};
  #pragma unroll
  for (int j = 0; j < 4; ++j) {
    int c = cq + j;
    float mean = stats[c] / cnt;
    float var  = stats[16 + c] / cnt - mean * mean;
    float rs   = rsqrtf(var + BNEPS);
    float t    = (y[j] - mean) * rs * gamma[c] + beta[c];
    y[j] = valid ? fmaxf(t, 0.0f) : 0.0f;
  }
  ((float4*)data)[i] = make_float4(y[0], y[1], y[2], y[3]);
}

// ---------------------------------------------------------------------------
// Launch: init -> scatter -> [conv -> bn] x2.  Workspace layout:
//   [0, ~8MB)        idx_map  (B * (64^3+1) int32, sentinel per batch)
//   [.., +33.5MB)    tmp1     (B * N * 16 fp32, layer-1 activations)
//   [.., +256B)      stats    (layer0: sum[16],sumsq[16]; layer1: same)
//   [.., +64B)       zrow     (16 fp32 zeros, branchless invalid-gather target)
// ---------------------------------------------------------------------------
extern "C" void kernel_launch(void* const* d_in, const int* in_sizes, int n_in,
                              void* d_out, int out_size, void* d_ws, size_t ws_size,
                              hipStream_t stream) {
  (void)in_sizes; (void)n_in; (void)out_size; (void)ws_size;
  const float* vf     = (const float*)d_in[0];
  const int*   xyz    = (const int*)d_in[1];
  const int*   nvalid = (const int*)d_in[2];
  const float* w0     = (const float*)d_in[3];
  const float* w1     = (const float*)d_in[4];
  const float* nw0    = (const float*)d_in[5];
  const float* nw1    = (const float*)d_in[6];
  const float* g0     = (const float*)d_in[7];
  const float* b0     = (const float*)d_in[8];
  const float* g1     = (const float*)d_in[9];
  const float* b1     = (const float*)d_in[10];
  float* out = (float*)d_out;

  char* ws = (char*)d_ws;
  size_t mapBytes = (size_t)BB * MAPSTRIDE * sizeof(int);
  mapBytes = (mapBytes + 255) & ~(size_t)255;                // 16B-align tmp1
  int*   imap  = (int*)ws;
  float* tmp1  = (float*)(ws + mapBytes);
  float* stats = (float*)(ws + mapBytes + (size_t)BB * NVOX * CIN * sizeof(float));
  float* zrow  = stats + 64;

  const int convBlocks = (BB * NVOX / 32) / 8;          // 2 tiles/wave, 8 waves/block
  const int bnBlocks   = (BB * NVOX * 4 + 255) / 256;

  sc3d_init<<<(BB * MAPSTRIDE + 255) / 256, 256, 0, stream>>>(imap, stats);
  sc3d_scatter<<<(BB * NVOX + 255) / 256, 256, 0, stream>>>(imap, xyz, nvalid);

  sc3d_conv<<<convBlocks, 256, 0, stream>>>(vf, xyz, nvalid, w0, nw0, imap, zrow, tmp1, stats);
  sc3d_bn<<<bnBlocks, 256, 0, stream>>>(tmp1, nvalid, g0, b0, stats);

  sc3d_conv<<<convBlocks, 256, 0, stream>>>(tmp1, xyz, nvalid, w1, nw1, imap, zrow, out, stats + 32);
  sc3d_bn<<<bnBlocks, 256, 0, stream>>>(out, nvalid, g1, b1, stats + 32);
}